// GAT_GCN_3925600108908
// MI455X (gfx1250) — compile-verified
//
#include <hip/hip_runtime.h>
#include <cstdint>
#include <cstddef>

// ---------------------------------------------------------------------------
// MI455X (gfx1250) implementation.
// Compute-bound: ~57 GFLOP of [64,K]x[K,N] f32 GEMMs -> V_WMMA_F32_16X16X4_F32.
// Weights (660KB) + points2 (67MB) are L2-resident (192MB L2), HBM traffic is
// negligible at 23.3 TB/s, so everything is fused per-query in LDS.
// ---------------------------------------------------------------------------

typedef __attribute__((ext_vector_type(2))) float v2f;
typedef __attribute__((ext_vector_type(8))) float v8f;

#define NSAMP 64
#define NQ    21      // N1
#define NP    1024    // N2
#define DCH   128     // D
#define CIN   131
#define KPAD  132     // CIN padded to multiple of 4
#define STR   132     // LDS row stride in floats (132 % 64 == 4 -> conflict free)
#define NB    128     // batch

__device__ __forceinline__ float lrelu(float x) { return x > 0.f ? x : 0.1f * x; }

// ---------------- top-64 nearest neighbors -------------------------------
__global__ void topk_kernel(const float* __restrict__ xyz1,
                            const float* __restrict__ xyz2,
                            int* __restrict__ idxg) {
  const int n1 = blockIdx.x, b = blockIdx.y, tid = threadIdx.x;
  __shared__ unsigned int keys[NP];
  __shared__ unsigned long long red[256];
  __shared__ int chosen[NSAMP];
  __shared__ float x1s[4];
  if (tid < 3) x1s[tid] = xyz1[((size_t)b * 3 + tid) * NQ + n1];
  __syncthreads();
  const float xa = x1s[0], xb = x1s[1], xc = x1s[2];
  for (int j = tid; j < NP; j += 256) {
    float dx = xyz2[((size_t)b * 3 + 0) * NP + j] - xa;
    float dy = xyz2[((size_t)b * 3 + 1) * NP + j] - xb;
    float dz = xyz2[((size_t)b * 3 + 2) * NP + j] - xc;
    float d = dx * dx + dy * dy + dz * dz;
    unsigned int bits = __float_as_uint(d);
    // float -> sortable unsigned key
    unsigned int key = (bits & 0x80000000u) ? ~bits : (bits | 0x80000000u);
    keys[j] = key;
  }
  __syncthreads();
  for (int it = 0; it < NSAMP; ++it) {
    unsigned long long best = ~0ull;
    #pragma unroll
    for (int t = 0; t < 4; ++t) {
      int j = tid + t * 256;
      unsigned long long p = ((unsigned long long)keys[j] << 32) | (unsigned int)j;
      best = p < best ? p : best;
    }
    red[tid] = best;
    __syncthreads();
    for (int s = 128; s > 0; s >>= 1) {
      if (tid < s) red[tid] = red[tid + s] < red[tid] ? red[tid + s] : red[tid];
      __syncthreads();
    }
    if (tid == 0) {
      int j = (int)(red[0] & 0xFFFFFFFFull);
      chosen[it] = j;
      keys[j] = 0xFFFFFFFFu;  // remove winner
    }
    __syncthreads();
  }
  if (tid < NSAMP) idxg[((size_t)b * NQ + n1) * NSAMP + tid] = chosen[tid];
}

// ---------------- graph branch: p1g, fq/fk projections --------------------
__global__ void graph_kernel(const float* __restrict__ points1,
                             const float* __restrict__ graph_a,
                             const float* __restrict__ gw_w,
                             const float* __restrict__ gw_b,
                             const float* __restrict__ fq_w,
                             const float* __restrict__ fk_w,
                             float* __restrict__ p1g_g,
                             float* __restrict__ fqv_g,
                             float* __restrict__ fkv_g) {
  const int b = blockIdx.x, tid = threadIdx.x;
  __shared__ float pts[DCH * NQ], tmp[DCH * NQ], p1g[DCH * NQ];
  for (int e = tid; e < DCH * NQ; e += 256) pts[e] = points1[(size_t)b * DCH * NQ + e];
  __syncthreads();
  // tmp[c,m] = sum_n pts[c,n] * graph_a[c,n,m]
  for (int e = tid; e < DCH * NQ; e += 256) {
    int c = e / NQ, m = e % NQ;
    float a = 0.f;
    for (int n = 0; n < NQ; ++n) a += pts[c * NQ + n] * graph_a[((size_t)c * NQ + n) * NQ + m];
    tmp[e] = a;
  }
  __syncthreads();
  // p1g[o,m] = lrelu(sum_c gw_w[o,c]*tmp[c,m] + gw_b[o])
  for (int e = tid; e < DCH * NQ; e += 256) {
    int o = e / NQ, m = e % NQ;
    float a = gw_b[o];
    const float* wr = gw_w + (size_t)o * DCH;
    for (int c = 0; c < DCH; ++c) a += wr[c] * tmp[c * NQ + m];
    a = lrelu(a);
    p1g[e] = a;
    p1g_g[(size_t)b * DCH * NQ + e] = a;
  }
  __syncthreads();
  for (int e = tid; e < DCH * NQ; e += 256) {
    int o = e / NQ, m = e % NQ;
    float aq = 0.f, ak = 0.f;
    const float* wq = fq_w + (size_t)o * DCH;
    const float* wk = fk_w + (size_t)o * DCH;
    for (int c = 0; c < DCH; ++c) {
      float p = p1g[c * NQ + m];
      aq += wq[c] * p;
      ak += wk[c] * p;
    }
    fqv_g[(size_t)b * DCH * NQ + e] = aq;
    fkv_g[(size_t)b * DCH * NQ + e] = ak;
  }
}

// ---------------- weight packing: Wp[p][n] = (W[n][2p], W[n][2p+1]) -------
__global__ void pack_w(const float* __restrict__ W, v2f* __restrict__ Wp,
                       int K, int Khalf, int N) {
  int i = blockIdx.x * blockDim.x + threadIdx.x;
  if (i >= Khalf * N) return;
  int p = i / N, n = i % N;
  int k0 = 2 * p, k1 = 2 * p + 1;
  v2f v;
  v.x = (k0 < K) ? W[(size_t)n * K + k0] : 0.f;
  v.y = (k1 < K) ? W[(size_t)n * K + k1] : 0.f;
  Wp[(size_t)p * N + n] = v;
}

// ---------------- fused main kernel ---------------------------------------
#define EP_NONE 0
#define EP_LB   1
#define EP_LBV  2
#define EP_SIG  3
#define EP_GATE 4

struct MMArgs {
  const float* A;       // LDS [64][STR]
  const v2f* Wp;        // packed B operand [Khalf][N]
  int K;                // 128 or 132
  int N;                // 128 or 256
  float* Out;           // LDS [64][STR]
  int ep;
  const float* bias;    // global, len N
  const float* vec;     // LDS len 128 (EP_LBV)
  const float* k2v;     // LDS len 256 (EP_SIG)
  float* Out2;          // EP_SIG second half
  const float* ascale;  // LDS len >= K, scales A columns (or null)
  const float* g1;      // LDS [64][STR] (EP_GATE)
  const float* p1e;     // LDS [64][STR] (EP_GATE)
};

// OUT[64][N] = IN[64][K] @ B  (B from packed weights), f32 WMMA 16x16x4.
// 8 waves; each wave owns col-tiles (w, w+8, ...) and all 4 row tiles so the
// B fragment is loaded once per k-step and reused across the 4 row tiles.
__device__ __forceinline__ void mm64(const MMArgs& p) {
  const int tid = threadIdx.x;
  const int wave = tid >> 5;
  const int lane = tid & 31;
  const int lhi = lane >> 4;   // 0: K=k,k+1  1: K=k+2,k+3 / C rows +8
  const int l15 = lane & 15;
  const int ntiles = p.N >> 4;
  for (int ct = wave; ct < ntiles; ct += 8) {
    const int n0 = ct * 16 + l15;
    v8f acc[4];
    #pragma unroll
    for (int r = 0; r < 4; ++r)
      acc[r] = (v8f){0.f, 0.f, 0.f, 0.f, 0.f, 0.f, 0.f, 0.f};
    for (int k = 0; k < p.K; k += 4) {
      v2f bf = p.Wp[(size_t)((k >> 1) + lhi) * p.N + n0];
      const int ka = k + 2 * lhi;
      const float* ab = p.A + ka;
      v2f f0 = *(const v2f*)(ab + (0 * 16 + l15) * STR);
      v2f f1 = *(const v2f*)(ab + (1 * 16 + l15) * STR);
      v2f f2 = *(const v2f*)(ab + (2 * 16 + l15) * STR);
      v2f f3 = *(const v2f*)(ab + (3 * 16 + l15) * STR);
      if (p.ascale) {
        v2f sc = *(const v2f*)(p.ascale + ka);
        f0 *= sc; f1 *= sc; f2 *= sc; f3 *= sc;
      }
      acc[0] = __builtin_amdgcn_wmma_f32_16x16x4_f32(false, f0, false, bf, (short)0, acc[0], false, false);
      acc[1] = __builtin_amdgcn_wmma_f32_16x16x4_f32(false, f1, false, bf, (short)0, acc[1], false, false);
      acc[2] = __builtin_amdgcn_wmma_f32_16x16x4_f32(false, f2, false, bf, (short)0, acc[2], false, false);
      acc[3] = __builtin_amdgcn_wmma_f32_16x16x4_f32(false, f3, false, bf, (short)0, acc[3], false, false);
    }
    // epilogue + store (C layout: vgpr j -> M=j(+8 for hi lanes), N=lane&15)
    #pragma unroll
    for (int r = 0; r < 4; ++r) {
      #pragma unroll
      for (int j = 0; j < 8; ++j) {
        const int row = r * 16 + j + 8 * lhi;
        const int col = n0;
        float x = acc[r][j];
        if (p.ep == EP_NONE) {
          p.Out[row * STR + col] = x;
        } else if (p.ep == EP_LB) {
          p.Out[row * STR + col] = lrelu(x + p.bias[col]);
        } else if (p.ep == EP_LBV) {
          p.Out[row * STR + col] = lrelu(x + p.bias[col] + p.vec[col]);
        } else if (p.ep == EP_SIG) {
          float y = x + p.bias[col];
          float a = 1.f / (1.f + expf(-p.k2v[col] * y));
          if (col < 128) p.Out[row * STR + col] = a;
          else p.Out2[row * STR + (col - 128)] = a;
        } else {  // EP_GATE
          float y = x + p.bias[col];
          y = y * p.g1[row * STR + col] + p.p1e[row * STR + col];
          p.Out[row * STR + col] = lrelu(y);
        }
      }
    }
  }
}

struct __align__(16) SM {
  float b0[64 * STR];
  float b1[64 * STR];
  float b2[64 * STR];
  float b3[64 * STR];
  float p1gv[DCH];
  float fqvv[DCH];
  float fkvv[DCH];
  float pts1v[DCH];
  float kmaxv[DCH];
  float vresv[DCH];
  float vmaxv[DCH];
  float k2v[2 * DCH];
  float x1v[4];
  int idxs[NSAMP];
};

__global__ void __launch_bounds__(256)
main_kernel(const float* __restrict__ xyz1, const float* __restrict__ xyz2,
            const float* __restrict__ points1, const float* __restrict__ points2,
            const int* __restrict__ idxg,
            const float* __restrict__ p1g_g, const float* __restrict__ fqv_g,
            const float* __restrict__ fkv_g,
            const v2f* __restrict__ qw0p, const float* __restrict__ qb0,
            const v2f* __restrict__ qw1p, const float* __restrict__ qb1,
            const v2f* __restrict__ qw2p, const float* __restrict__ qb2,
            const v2f* __restrict__ kw0p, const float* __restrict__ kb0,
            const v2f* __restrict__ kw1p, const float* __restrict__ kb1,
            const float* __restrict__ kw2, const float* __restrict__ kb2,
            const v2f* __restrict__ fv1p, const v2f* __restrict__ fv2p,
            const v2f* __restrict__ vw0p, const float* __restrict__ vb0,
            const v2f* __restrict__ vw1p, const float* __restrict__ vb1,
            const float* __restrict__ vw2, const float* __restrict__ vb2,
            float* __restrict__ out) {
  const int n1 = blockIdx.x, b = blockIdx.y, tid = threadIdx.x;
  __shared__ SM sm;

  // ---- gather inputs ----
  if (tid < NSAMP) sm.idxs[tid] = idxg[((size_t)b * NQ + n1) * NSAMP + tid];
  if (tid < 3) sm.x1v[tid] = xyz1[((size_t)b * 3 + tid) * NQ + n1];
  if (tid >= 64 && tid < 192) {
    int c = tid - 64;
    size_t base = ((size_t)b * DCH + c) * NQ + n1;
    sm.p1gv[c] = p1g_g[base];
    sm.fqvv[c] = fqv_g[base];
    sm.fkvv[c] = fkv_g[base];
    sm.pts1v[c] = points1[base];
  }
  __syncthreads();
  for (int e = tid; e < NSAMP * STR; e += 256) {
    int s = e / STR, c = e % STR;
    int j = sm.idxs[s];
    float val;
    if (c < DCH) val = points2[((size_t)b * DCH + c) * NP + j];
    else if (c < CIN) { int d = c - DCH; val = xyz2[((size_t)b * 3 + d) * NP + j] - sm.x1v[d]; }
    else val = 0.f;
    sm.b0[s * STR + c] = val;  // X = new_points [64][131] (+pad)
  }
  __syncthreads();

  MMArgs m{};
  // ---- k path ----
  m.A = sm.b0; m.Wp = kw0p; m.K = KPAD; m.N = 128; m.Out = sm.b1;
  m.ep = EP_LBV; m.bias = kb0; m.vec = sm.fkvv; m.ascale = nullptr;
  mm64(m); __syncthreads();
  m.A = sm.b1; m.Wp = kw1p; m.K = 128; m.Out = sm.b2; m.ep = EP_LB; m.bias = kb1;
  mm64(m); __syncthreads();
  if (tid < DCH) {
    float mx = -3.4e38f;
    for (int s = 0; s < NSAMP; ++s) mx = fmaxf(mx, sm.b2[s * STR + tid]);
    sm.kmaxv[tid] = mx;
  }
  __syncthreads();
  {
    int o = tid;
    if (o < 2 * DCH) {
      float a = kb2[o];
      const float* wr = kw2 + (size_t)o * DCH;
      for (int c = 0; c < DCH; ++c) a += wr[c] * sm.kmaxv[c];
      sm.k2v[o] = a;
    }
  }
  __syncthreads();

  // ---- q path ----
  m.A = sm.b0; m.Wp = qw0p; m.K = KPAD; m.Out = sm.b1;
  m.ep = EP_LBV; m.bias = qb0; m.vec = sm.fqvv;
  mm64(m); __syncthreads();
  m.A = sm.b1; m.Wp = qw1p; m.K = 128; m.Out = sm.b2; m.ep = EP_LB; m.bias = qb1;
  mm64(m); __syncthreads();

  // ---- v0 = X @ fv1 (last use of X) ----
  m.A = sm.b0; m.Wp = fv1p; m.K = KPAD; m.Out = sm.b1; m.ep = EP_NONE; m.bias = nullptr;
  mm64(m); __syncthreads();

  // ---- gates: a = sigmoid(k2 * (q1@qw2+qb2)); g1 -> b0, g2 -> b3 ----
  m.A = sm.b2; m.Wp = qw2p; m.K = 128; m.N = 256; m.Out = sm.b0; m.Out2 = sm.b3;
  m.ep = EP_SIG; m.bias = qb2; m.k2v = sm.k2v;
  mm64(m); __syncthreads();

  // ---- p1e = (g2 * p1g) @ fv2 -> b2 ----
  m.A = sm.b3; m.Wp = fv2p; m.K = 128; m.N = 128; m.Out = sm.b2;
  m.ep = EP_NONE; m.bias = nullptr; m.ascale = sm.p1gv;
  mm64(m); m.ascale = nullptr; __syncthreads();

  // ---- v = lrelu(v0*g1 + p1e) + points1 -> b3 ; v_res = mean ----
  for (int e = tid; e < NSAMP * DCH; e += 256) {
    int s = e >> 7, c = e & 127;
    float y = sm.b1[s * STR + c] * sm.b0[s * STR + c] + sm.b2[s * STR + c];
    sm.b3[s * STR + c] = lrelu(y) + sm.pts1v[c];
  }
  __syncthreads();
  if (tid < DCH) {
    float a = 0.f;
    for (int s = 0; s < NSAMP; ++s) a += sm.b3[s * STR + tid];
    sm.vresv[tid] = a * (1.f / NSAMP);
  }
  __syncthreads();

  // ---- t = lrelu((v@vw0+vb0)*g1 + p1e) -> b1 ----
  m.A = sm.b3; m.Wp = vw0p; m.Out = sm.b1; m.ep = EP_GATE;
  m.bias = vb0; m.g1 = sm.b0; m.p1e = sm.b2;
  mm64(m); __syncthreads();

  // ---- t2 = lrelu(t@vw1+vb1) -> b3 ----
  m.A = sm.b1; m.Wp = vw1p; m.Out = sm.b3; m.ep = EP_LB; m.bias = vb1;
  mm64(m); __syncthreads();

  // ---- vmax, final conv, output ----
  if (tid < DCH) {
    float mx = -3.4e38f;
    for (int s = 0; s < NSAMP; ++s) mx = fmaxf(mx, sm.b3[s * STR + tid]);
    sm.vmaxv[tid] = mx;
  }
  __syncthreads();
  if (tid < DCH) {
    float a = vb2[tid];
    const float* wr = vw2 + (size_t)tid * DCH;
    for (int c = 0; c < DCH; ++c) a += wr[c] * sm.vmaxv[c];
    out[((size_t)b * DCH + tid) * NQ + n1] = lrelu(a) + sm.vresv[tid];
  }
}

// ---------------------------------------------------------------------------
extern "C" void kernel_launch(void* const* d_in, const int* in_sizes, int n_in,
                              void* d_out, int out_size, void* d_ws, size_t ws_size,
                              hipStream_t stream) {
  (void)in_sizes; (void)n_in; (void)out_size; (void)ws_size;
  const float* xyz1    = (const float*)d_in[0];
  const float* xyz2    = (const float*)d_in[1];
  const float* points1 = (const float*)d_in[2];
  const float* points2 = (const float*)d_in[3];
  const float* graph_a = (const float*)d_in[4];
  const float* gw_w = (const float*)d_in[5];
  const float* gw_b = (const float*)d_in[6];
  const float* fq_w = (const float*)d_in[7];
  const float* fk_w = (const float*)d_in[8];
  const float* fv1w = (const float*)d_in[9];
  const float* fv2w = (const float*)d_in[10];
  const float* qw0 = (const float*)d_in[11]; const float* qb0 = (const float*)d_in[12];
  const float* qw1 = (const float*)d_in[13]; const float* qb1 = (const float*)d_in[14];
  const float* qw2 = (const float*)d_in[15]; const float* qb2 = (const float*)d_in[16];
  const float* kw0 = (const float*)d_in[17]; const float* kb0 = (const float*)d_in[18];
  const float* kw1 = (const float*)d_in[19]; const float* kb1 = (const float*)d_in[20];
  const float* kw2 = (const float*)d_in[21]; const float* kb2 = (const float*)d_in[22];
  const float* vw0 = (const float*)d_in[23]; const float* vb0 = (const float*)d_in[24];
  const float* vw1 = (const float*)d_in[25]; const float* vb1 = (const float*)d_in[26];
  const float* vw2 = (const float*)d_in[27]; const float* vb2 = (const float*)d_in[28];
  float* out = (float*)d_out;

  char* ws = (char*)d_ws;
  size_t off = 0;
  int* idxg = (int*)(ws + off);      off += (size_t)NB * NQ * NSAMP * sizeof(int);
  float* p1g_g = (float*)(ws + off); off += (size_t)NB * DCH * NQ * sizeof(float);
  float* fqv_g = (float*)(ws + off); off += (size_t)NB * DCH * NQ * sizeof(float);
  float* fkv_g = (float*)(ws + off); off += (size_t)NB * DCH * NQ * sizeof(float);
  v2f* qw0p = (v2f*)(ws + off); off += (size_t)66 * 128 * sizeof(v2f);
  v2f* kw0p = (v2f*)(ws + off); off += (size_t)66 * 128 * sizeof(v2f);
  v2f* fv1p = (v2f*)(ws + off); off += (size_t)66 * 128 * sizeof(v2f);
  v2f* qw1p = (v2f*)(ws + off); off += (size_t)64 * 128 * sizeof(v2f);
  v2f* kw1p = (v2f*)(ws + off); off += (size_t)64 * 128 * sizeof(v2f);
  v2f* fv2p = (v2f*)(ws + off); off += (size_t)64 * 128 * sizeof(v2f);
  v2f* vw0p = (v2f*)(ws + off); off += (size_t)64 * 128 * sizeof(v2f);
  v2f* vw1p = (v2f*)(ws + off); off += (size_t)64 * 128 * sizeof(v2f);
  v2f* qw2p = (v2f*)(ws + off); off += (size_t)64 * 256 * sizeof(v2f);

  topk_kernel<<<dim3(NQ, NB), 256, 0, stream>>>(xyz1, xyz2, idxg);
  graph_kernel<<<NB, 256, 0, stream>>>(points1, graph_a, gw_w, gw_b, fq_w, fk_w,
                                       p1g_g, fqv_g, fkv_g);
  auto packN = [&](const float* W, v2f* Wp, int K, int Khalf, int N) {
    int tot = Khalf * N;
    pack_w<<<(tot + 255) / 256, 256, 0, stream>>>(W, Wp, K, Khalf, N);
  };
  packN(qw0, qw0p, CIN, 66, 128);
  packN(kw0, kw0p, CIN, 66, 128);
  packN(fv1w, fv1p, CIN, 66, 128);
  packN(qw1, qw1p, 128, 64, 128);
  packN(kw1, kw1p, 128, 64, 128);
  packN(fv2w, fv2p, 128, 64, 128);
  packN(vw0, vw0p, 128, 64, 128);
  packN(vw1, vw1p, 128, 64, 128);
  packN(qw2, qw2p, 128, 64, 256);

  main_kernel<<<dim3(NQ, NB), 256, 0, stream>>>(
      xyz1, xyz2, points1, points2, idxg, p1g_g, fqv_g, fkv_g,
      qw0p, qb0, qw1p, qb1, qw2p, qb2,
      kw0p, kb0, kw1p, kb1, kw2, kb2,
      fv1p, fv2p, vw0p, vb0, vw1p, vb1, vw2, vb2, out);
}